// DendriNet_63428077027564
// MI455X (gfx1250) — compile-verified
//
#include <hip/hip_runtime.h>
#include <cstdint>

// ---------------------------------------------------------------------------
// DendriNet forward, MI455X (gfx1250).
// Pass 1: exact top-32 radix-select per pre-weight row -> packed (idx|fp16 w).
// Pass 2: fused sparse gather forward; x rows staged via TDM
//         (tensor_load_to_lds, TENSORcnt), inh rows via per-lane async
//         global->LDS copies (ASYNCcnt). All gathers hit LDS.
// ---------------------------------------------------------------------------

#define D_IN      4096
#define NROWS     5376     // 2048+2048+512+512+128+128
#define OFF_E0    0
#define OFF_I0    2048
#define OFF_E1    4096
#define OFF_I1    4608
#define OFF_ES    5120
#define OFF_IS    5248
#define BT        4        // batch rows per workgroup in pass 2

typedef __attribute__((ext_vector_type(4))) unsigned v4u;
typedef __attribute__((ext_vector_type(8))) int      v8i;
typedef __attribute__((ext_vector_type(4))) int      v4i;

#if defined(__has_builtin)
# if __has_builtin(__builtin_amdgcn_tensor_load_to_lds) && __has_builtin(__builtin_amdgcn_s_wait_tensorcnt)
#  define HAVE_TDM 1
# endif
#endif

// ---------------- gfx1250 async global->LDS helpers ------------------------
__device__ __forceinline__ void async_ld_b128(unsigned lds_off, const void* gptr) {
    unsigned long long ga = (unsigned long long)(uintptr_t)gptr;
    asm volatile("global_load_async_to_lds_b128 %0, %1, off"
                 :: "v"(lds_off), "v"(ga) : "memory");
}
__device__ __forceinline__ void wait_async0() {
    asm volatile("s_wait_asynccnt 0" ::: "memory");
}

// One-shot TDM DMA of `nelem` contiguous f32 from global to LDS (per-wave op).
__device__ __forceinline__ void tdm_copy_f32(unsigned lds_off, const void* gptr,
                                             unsigned nelem) {
#if defined(HAVE_TDM)
    unsigned long long ga = (unsigned long long)(uintptr_t)gptr;
    v4u g0;
    g0.x = 1u;                                        // count=1 valid descriptor
    g0.y = lds_off;                                   // lds_addr (bytes)
    g0.z = (unsigned)ga;                              // global_addr[31:0]
    g0.w = (unsigned)(ga >> 32) | (2u << 30);         // global_addr[56:32] | type=2
    v8i g1;
    g1[0] = 0x00020000;                               // data_size=2 (4B); no mask/pad/iter
    g1[1] = (int)((nelem & 0xFFFFu) << 16);           // tensor_dim0[15:0]
    g1[2] = (int)(nelem >> 16);                       // tensor_dim0[31:16]; tensor_dim1=0
    g1[3] = (int)((nelem & 0xFFFFu) << 16);           // tile_dim0 (16-bit)
    g1[4] = 0;                                        // tile_dim1/2 unused
    g1[5] = (int)nelem;                               // tensor_dim0_stride[31:0]
    g1[6] = 0;                                        // stride[47:32]; dim1_stride lo
    g1[7] = 0;
    v4i z4 = {0, 0, 0, 0};
    v8i z8 = {0, 0, 0, 0, 0, 0, 0, 0};
    __builtin_amdgcn_tensor_load_to_lds(g0, g1, z4, z4, z8, 0);
    __builtin_amdgcn_s_wait_tensorcnt(0);
#else
    (void)lds_off; (void)gptr; (void)nelem;
#endif
}

__device__ __forceinline__ float h2f(unsigned e) {
    _Float16 h = __builtin_bit_cast(_Float16, (unsigned short)(e & 0xFFFFu));
    return (float)h;
}

// ---------------------------------------------------------------------------
// Kernel 1: per-row exact top-32 (radix select over monotonic keys in LDS)
// ---------------------------------------------------------------------------
__global__ __launch_bounds__(256) void dn_topk(
    const float* __restrict__ e0, const float* __restrict__ i0,
    const float* __restrict__ e1, const float* __restrict__ i1,
    const float* __restrict__ es, const float* __restrict__ is_,
    uint32_t* __restrict__ packed)
{
    __shared__ unsigned keys[D_IN];
    __shared__ unsigned hist[4096];
    __shared__ unsigned csum[256];
    __shared__ unsigned s_prefix, s_kneed, s_c1, s_c2;

    const int t   = threadIdx.x;
    const int row = blockIdx.x;

    const float* src;
    if      (row < OFF_I0) src = e0 + (size_t)(row - OFF_E0) * D_IN;
    else if (row < OFF_E1) src = i0 + (size_t)(row - OFF_I0) * D_IN;
    else if (row < OFF_I1) src = e1 + (size_t)(row - OFF_E1) * D_IN;
    else if (row < OFF_ES) src = i1 + (size_t)(row - OFF_I1) * D_IN;
    else if (row < OFF_IS) src = es + (size_t)(row - OFF_ES) * D_IN;
    else                   src = is_ + (size_t)(row - OFF_IS) * D_IN;

    // monotonic key: ascending uint == ascending float
    #pragma unroll
    for (int j = 0; j < 16; ++j) {
        int i = t + j * 256;
        unsigned b = __float_as_uint(src[i]);
        keys[i] = (b >> 31) ? ~b : (b | 0x80000000u);
    }
    if (t == 0) { s_prefix = 0u; s_kneed = 32u; s_c1 = 0u; s_c2 = 0u; }
    __syncthreads();

    const int      shifts[3] = {20, 8, 0};
    const int      nbins [3] = {4096, 4096, 256};
    const unsigned pmsk  [3] = {0u, 0xFFF00000u, 0xFFFFFF00u};

    for (int p = 0; p < 3; ++p) {
        const int      sh = shifts[p];
        const int      nb = nbins[p];
        const unsigned pm = pmsk[p];
        const unsigned pref = s_prefix;
        const unsigned kn   = s_kneed;

        for (int j = t; j < nb; j += 256) hist[j] = 0u;
        __syncthreads();

        #pragma unroll
        for (int j = 0; j < 16; ++j) {
            unsigned key = keys[t + j * 256];
            if ((key & pm) == pref)
                atomicAdd(&hist[(key >> sh) & (unsigned)(nb - 1)], 1u);
        }
        __syncthreads();

        const int nch = nb / 16;
        if (t < nch) {
            unsigned s = 0;
            #pragma unroll
            for (int j = 0; j < 16; ++j) s += hist[t * 16 + j];
            csum[t] = s;
        }
        __syncthreads();

        if (t == 0) {
            unsigned cum = 0; int ch = 0;
            for (int c = nch - 1; c >= 0; --c) {
                if (cum + csum[c] >= kn) { ch = c; break; }
                cum += csum[c];
            }
            int bin = ch * 16;
            for (int b = ch * 16 + 15; b >= ch * 16; --b) {
                if (cum + hist[b] >= kn) { bin = b; break; }
                cum += hist[b];
            }
            s_prefix = pref | ((unsigned)bin << sh);
            s_kneed  = kn - cum;           // still needed from the == bin
        }
        __syncthreads();
    }

    const unsigned T     = s_prefix;      // exact 32nd-largest key
    const unsigned kneed = s_kneed;       // how many to take from key == T
    const unsigned base  = (unsigned)row * 32u;

    #pragma unroll
    for (int j = 0; j < 16; ++j) {
        int i = t + j * 256;
        unsigned key = keys[i];
        int slot = -1;
        if (key > T) {
            slot = (int)atomicAdd(&s_c1, 1u);           // strictly greater: 32-kneed of them
        } else if (key == T) {
            unsigned q = atomicAdd(&s_c2, 1u);
            if (q < kneed) slot = (int)(32u - kneed + q);
        }
        if (slot >= 0) {
            unsigned b = (key & 0x80000000u) ? (key ^ 0x80000000u) : ~key;
            float w = __expf(__uint_as_float(b));
            _Float16 h = (_Float16)w;
            unsigned short hb = __builtin_bit_cast(unsigned short, h);
            packed[base + (unsigned)slot] = ((unsigned)i << 16) | (unsigned)hb;
        }
    }
}

// ---------------------------------------------------------------------------
// Kernel 2: fused forward over a BT-row batch tile, LDS-resident gathers
// ---------------------------------------------------------------------------
__global__ __launch_bounds__(512) void dn_forward(
    const float* __restrict__ x, const float* __restrict__ inh,
    const uint32_t* __restrict__ packed,
    const float* __restrict__ bw1, const float* __restrict__ bws,
    const float* __restrict__ vth_ps, const float* __restrict__ lam,
    float* __restrict__ out)
{
    extern __shared__ float smem[];
    float* xs = smem;                    // BT * 4096
    float* hs = xs + BT * D_IN;          // BT * 4096 (inhibitory)
    float* h0 = hs + BT * D_IN;          // BT * 2048
    float* h1 = h0 + BT * 2048;          // BT * 512

    const int t  = threadIdx.x;
    const int b0 = blockIdx.x * BT;

    // ---- Phase A: stage x (TDM DMA) and inh (async copies) into LDS -------
    {
        const float* gx = x   + (size_t)b0 * D_IN;
        const float* gi = inh + (size_t)b0 * D_IN;
        const unsigned xoff = (unsigned)(uintptr_t)xs;
        const unsigned ioff = (unsigned)(uintptr_t)hs;
#if defined(HAVE_TDM)
        if (t < 32) {                     // one TDM DMA per workgroup (wave 0)
            tdm_copy_f32(xoff, gx, BT * D_IN);
        }
#else
        for (int c = t; c < (BT * D_IN) / 4; c += 512)
            async_ld_b128(xoff + (unsigned)c * 16u, (const void*)(gx + c * 4));
#endif
        for (int c = t; c < (BT * D_IN) / 4; c += 512)    // 16B chunks
            async_ld_b128(ioff + (unsigned)c * 16u, (const void*)(gi + c * 4));
        wait_async0();
    }
    __syncthreads();

    // ---- Phase B: layer 0 (2048 branches, no branch input) -----------------
    for (int oi = 0; oi < 2048; oi += 512) {
        const int o = oi + t;
        const uint32_t* pe = packed + (size_t)(OFF_E0 + o) * 32;
        const uint32_t* pi = packed + (size_t)(OFF_I0 + o) * 32;
        float e[BT] = {0.f, 0.f, 0.f, 0.f};
        float ih[BT] = {0.f, 0.f, 0.f, 0.f};
        #pragma unroll 2
        for (int k = 0; k < 32; k += 4) {
            uint4 pk = *(const uint4*)(pe + k);
            unsigned en[4] = {pk.x, pk.y, pk.z, pk.w};
            #pragma unroll
            for (int u = 0; u < 4; ++u) {
                const int idx = (int)(en[u] >> 16);
                const float w = h2f(en[u]);
                #pragma unroll
                for (int b = 0; b < BT; ++b) e[b] += w * xs[b * D_IN + idx];
            }
        }
        #pragma unroll 2
        for (int k = 0; k < 32; k += 4) {
            uint4 pk = *(const uint4*)(pi + k);
            unsigned en[4] = {pk.x, pk.y, pk.z, pk.w};
            #pragma unroll
            for (int u = 0; u < 4; ++u) {
                const int idx = (int)(en[u] >> 16);
                const float w = h2f(en[u]);
                #pragma unroll
                for (int b = 0; b < BT; ++b) ih[b] += w * hs[b * D_IN + idx];
            }
        }
        #pragma unroll
        for (int b = 0; b < BT; ++b)
            h0[b * 2048 + o] = e[b] / (e[b] + 1.0f + ih[b]);
    }
    __syncthreads();

    // ---- Phase C: layer 1 (512 branches, aggregates 4 children) -----------
    {
        const int o = t;  // blockDim == 512
        const uint32_t* pe = packed + (size_t)(OFF_E1 + o) * 32;
        const uint32_t* pi = packed + (size_t)(OFF_I1 + o) * 32;
        float e[BT] = {0.f, 0.f, 0.f, 0.f};
        float ih[BT] = {0.f, 0.f, 0.f, 0.f};
        #pragma unroll 2
        for (int k = 0; k < 32; k += 4) {
            uint4 pk = *(const uint4*)(pe + k);
            unsigned en[4] = {pk.x, pk.y, pk.z, pk.w};
            #pragma unroll
            for (int u = 0; u < 4; ++u) {
                const int idx = (int)(en[u] >> 16);
                const float w = h2f(en[u]);
                #pragma unroll
                for (int b = 0; b < BT; ++b) e[b] += w * xs[b * D_IN + idx];
            }
        }
        #pragma unroll 2
        for (int k = 0; k < 32; k += 4) {
            uint4 pk = *(const uint4*)(pi + k);
            unsigned en[4] = {pk.x, pk.y, pk.z, pk.w};
            #pragma unroll
            for (int u = 0; u < 4; ++u) {
                const int idx = (int)(en[u] >> 16);
                const float w = h2f(en[u]);
                #pragma unroll
                for (int b = 0; b < BT; ++b) ih[b] += w * hs[b * D_IN + idx];
            }
        }
        const float4 bw = *(const float4*)(bw1 + (size_t)o * 4);
        const float bwsum = bw.x + bw.y + bw.z + bw.w;
        #pragma unroll
        for (int b = 0; b < BT; ++b) {
            const float* hb = h0 + b * 2048 + o * 4;
            const float cur = hb[0] * bw.x + hb[1] * bw.y + hb[2] * bw.z + hb[3] * bw.w;
            h1[b * 512 + o] = (e[b] + cur) / (e[b] + 1.0f + bwsum + ih[b]);
        }
    }
    __syncthreads();

    // ---- Phase D: soma (128 outputs x BT batch = 512 tasks) ---------------
    {
        const int o = t & 127;
        const int b = t >> 7;
        const uint32_t* pe = packed + (size_t)(OFF_ES + o) * 32;
        const uint32_t* pi = packed + (size_t)(OFF_IS + o) * 32;
        float e = 0.f, ih = 0.f;
        #pragma unroll 2
        for (int k = 0; k < 32; k += 4) {
            uint4 pk = *(const uint4*)(pe + k);
            unsigned en[4] = {pk.x, pk.y, pk.z, pk.w};
            #pragma unroll
            for (int u = 0; u < 4; ++u)
                e += h2f(en[u]) * xs[b * D_IN + (int)(en[u] >> 16)];
        }
        #pragma unroll 2
        for (int k = 0; k < 32; k += 4) {
            uint4 pk = *(const uint4*)(pi + k);
            unsigned en[4] = {pk.x, pk.y, pk.z, pk.w};
            #pragma unroll
            for (int u = 0; u < 4; ++u)
                ih += h2f(en[u]) * hs[b * D_IN + (int)(en[u] >> 16)];
        }
        const float4 bw = *(const float4*)(bws + (size_t)o * 4);
        const float bwsum = bw.x + bw.y + bw.z + bw.w;
        const float* hb = h1 + b * 512 + o * 4;
        const float cur = hb[0] * bw.x + hb[1] * bw.y + hb[2] * bw.z + hb[3] * bw.w;
        const float v   = (e + cur) / (e + 1.0f + bwsum + ih);
        const float vth = 1.0f / (1.0f + __expf(-vth_ps[o]));
        const float vd  = v - vth;
        const float r   = __expf(lam[o]) * vd * vd;
        out[(size_t)(b0 + b) * 128 + o] = (vd < 0.0f) ? 0.0f : r;
    }
}

// ---------------------------------------------------------------------------
extern "C" void kernel_launch(void* const* d_in, const int* in_sizes, int n_in,
                              void* d_out, int out_size, void* d_ws, size_t ws_size,
                              hipStream_t stream) {
    (void)in_sizes; (void)n_in; (void)out_size; (void)ws_size;
    const float* x    = (const float*)d_in[0];
    const float* inh  = (const float*)d_in[1];
    const float* e0   = (const float*)d_in[2];
    const float* i0   = (const float*)d_in[3];
    const float* e1   = (const float*)d_in[4];
    const float* i1   = (const float*)d_in[5];
    const float* bw1  = (const float*)d_in[6];
    const float* es   = (const float*)d_in[7];
    const float* is_  = (const float*)d_in[8];
    const float* bws  = (const float*)d_in[9];
    const float* vth  = (const float*)d_in[10];
    const float* lam  = (const float*)d_in[11];
    float*    out     = (float*)d_out;
    uint32_t* packed  = (uint32_t*)d_ws;      // NROWS*32*4 = 688,128 bytes

    dn_topk<<<NROWS, 256, 0, stream>>>(e0, i0, e1, i1, es, is_, packed);

    const size_t smem = (size_t)(BT * D_IN * 2 + BT * 2048 + BT * 512) * sizeof(float); // 168 KB
    dn_forward<<<4096 / BT, 512, smem, stream>>>(x, inh, packed, bw1, bws, vth, lam, out);
}